// NodeEmbedding_53549652247107
// MI455X (gfx1250) — compile-verified
//
#include <hip/hip_runtime.h>
#include <math.h>

typedef __attribute__((ext_vector_type(2))) float v2f;
typedef __attribute__((ext_vector_type(8))) float v8f;

#define NN 100000
#define NE 1000000
#define DIM 64
#define BN_EPS 1e-5f
#define STATS_BLOCKS 512

__global__ void k_init_deg(float* __restrict__ deg) {
    int i = blockIdx.x * blockDim.x + threadIdx.x;
    if (i < NN) deg[i] = 1.0f;
}

__global__ void k_degree(const int* __restrict__ dst, float* __restrict__ deg) {
    int e = blockIdx.x * blockDim.x + threadIdx.x;
    if (e < NE) atomicAdd(&deg[dst[e]], 1.0f);
}

__global__ void k_dinv(const float* __restrict__ deg, float* __restrict__ dinv) {
    int i = blockIdx.x * blockDim.x + threadIdx.x;
    if (i < NN) dinv[i] = rsqrtf(deg[i]);   // deg >= 1 always (self loop)
}

// h = x @ W via V_WMMA_F32_16X16X4_F32; seeds agg(=d_out) with self-loop term.
__global__ void __launch_bounds__(128)
k_gemm(const float* __restrict__ x, const float* __restrict__ W,
       const float* __restrict__ dinv,
       float* __restrict__ h, float* __restrict__ agg) {
    const int wave = threadIdx.x >> 5;
    const int lane = threadIdx.x & 31;
    const int tile = blockIdx.x * (blockDim.x >> 5) + wave;
    if (tile >= NN / 16) return;              // wave-uniform: EXEC stays all-ones
    const int row_base = tile * 16;
    const int m   = lane & 15;
    const int sel = lane >> 4;

    v8f c0 = {}, c1 = {}, c2 = {}, c3 = {};
    const float* xrow = x + (size_t)(row_base + m) * DIM + 2 * sel;
    #pragma unroll
    for (int kk = 0; kk < 16; ++kk) {
        const int k = kk * 4;
        v2f a;
        a.x = xrow[k];
        a.y = xrow[k + 1];
        const float* wb = W + (size_t)(k + 2 * sel) * DIM + m;
        v2f b0, b1, b2, b3;
        b0.x = wb[0];  b0.y = wb[DIM];
        b1.x = wb[16]; b1.y = wb[DIM + 16];
        b2.x = wb[32]; b2.y = wb[DIM + 32];
        b3.x = wb[48]; b3.y = wb[DIM + 48];
        c0 = __builtin_amdgcn_wmma_f32_16x16x4_f32(false, a, false, b0, (short)0, c0, false, false);
        c1 = __builtin_amdgcn_wmma_f32_16x16x4_f32(false, a, false, b1, (short)0, c1, false, false);
        c2 = __builtin_amdgcn_wmma_f32_16x16x4_f32(false, a, false, b2, (short)0, c2, false, false);
        c3 = __builtin_amdgcn_wmma_f32_16x16x4_f32(false, a, false, b3, (short)0, c3, false, false);
    }
    float dv2[8];
    #pragma unroll
    for (int r = 0; r < 8; ++r) {
        float d = dinv[row_base + r + 8 * sel];
        dv2[r] = d * d;
    }
    #pragma unroll
    for (int r = 0; r < 8; ++r) {
        const size_t o = (size_t)(row_base + r + 8 * sel) * DIM + m;
        h[o]        = c0[r];
        h[o + 16]   = c1[r];
        h[o + 32]   = c2[r];
        h[o + 48]   = c3[r];
        agg[o]      = c0[r] * dv2[r];
        agg[o + 16] = c1[r] * dv2[r];
        agg[o + 32] = c2[r] * dv2[r];
        agg[o + 48] = c3[r] * dv2[r];
    }
}

__global__ void k_scatter(const int* __restrict__ src, const int* __restrict__ dst,
                          const float* __restrict__ dinv, const float* __restrict__ h,
                          float* __restrict__ agg) {
    unsigned gid = blockIdx.x * blockDim.x + threadIdx.x;
    unsigned e = gid >> 4;
    if (e >= NE) return;
    const int q = (gid & 15) * 4;
    const int s = src[e];
    const int d = dst[e];
    const float nrm = dinv[s] * dinv[d];
    const float4 hv = *(const float4*)(h + (size_t)s * DIM + q);
    float* ap = agg + (size_t)d * DIM + q;
    atomicAdd(ap + 0, hv.x * nrm);
    atomicAdd(ap + 1, hv.y * nrm);
    atomicAdd(ap + 2, hv.z * nrm);
    atomicAdd(ap + 3, hv.w * nrm);
}

__global__ void __launch_bounds__(256)
k_stats(const float* __restrict__ agg, const float* __restrict__ bias,
        float* __restrict__ partials) {
    const int f = threadIdx.x & 63;
    const int sub = threadIdx.x >> 6;
    const float bb = bias[f];
    float s = 0.f, ss = 0.f;
    for (int n = blockIdx.x * 4 + sub; n < NN; n += gridDim.x * 4) {
        float v = agg[(size_t)n * DIM + f] + bb;
        v = fmaxf(v, 0.f);
        s += v;
        ss += v * v;
    }
    __shared__ float rs[256];
    __shared__ float rq[256];
    rs[threadIdx.x] = s;
    rq[threadIdx.x] = ss;
    __syncthreads();
    if (sub == 0) {
        partials[blockIdx.x * 128 + f]      = rs[f] + rs[f + 64] + rs[f + 128] + rs[f + 192];
        partials[blockIdx.x * 128 + 64 + f] = rq[f] + rq[f + 64] + rq[f + 128] + rq[f + 192];
    }
}

__global__ void k_finalize(const float* __restrict__ partials,
                           const float* __restrict__ gamma, const float* __restrict__ beta,
                           float* __restrict__ sshift) {
    const int f = threadIdx.x;
    if (f >= 64) return;
    float s = 0.f, ss = 0.f;
    for (int b = 0; b < STATS_BLOCKS; ++b) {
        s  += partials[b * 128 + f];
        ss += partials[b * 128 + 64 + f];
    }
    const float invN = 1.0f / (float)NN;
    const float mean = s * invN;
    float var = ss * invN - mean * mean;
    var = fmaxf(var, 0.f);
    const float sc = gamma[f] * rsqrtf(var + BN_EPS);
    sshift[f]      = sc;
    sshift[64 + f] = beta[f] - mean * sc;
}

__global__ void k_apply(const float* __restrict__ bias, const float* __restrict__ sshift,
                        float* __restrict__ out) {
    size_t i = (size_t)blockIdx.x * blockDim.x + threadIdx.x;
    if (i >= (size_t)NN * DIM) return;
    const int f = (int)(i & 63);
    float v = fmaxf(out[i] + bias[f], 0.f);
    out[i] = v * sshift[f] + sshift[64 + f];
}

extern "C" void kernel_launch(void* const* d_in, const int* in_sizes, int n_in,
                              void* d_out, int out_size, void* d_ws, size_t ws_size,
                              hipStream_t stream) {
    (void)in_sizes; (void)n_in; (void)out_size; (void)ws_size;
    const float* x     = (const float*)d_in[0];
    const int*   ei    = (const int*)d_in[1];
    const float* W     = (const float*)d_in[2];
    const float* b     = (const float*)d_in[3];
    const float* gamma = (const float*)d_in[4];
    const float* beta  = (const float*)d_in[5];
    float* out = (float*)d_out;                     // doubles as agg accumulator

    const int* src  = ei;
    const int* dstp = ei + NE;

    char* ws = (char*)d_ws;
    float* deg      = (float*)ws;                                   // NN floats
    float* dinv     = (float*)(ws + 400000);                        // NN floats
    float* h        = (float*)(ws + 800000);                        // NN*DIM floats
    float* partials = (float*)(ws + 800000 + (size_t)NN * DIM * 4); // 512*128 floats
    float* sshift   = partials + STATS_BLOCKS * 128;                // 128 floats

    k_init_deg<<<(NN + 255) / 256, 256, 0, stream>>>(deg);
    k_degree  <<<(NE + 255) / 256, 256, 0, stream>>>(dstp, deg);
    k_dinv    <<<(NN + 255) / 256, 256, 0, stream>>>(deg, dinv);

    const int tiles = NN / 16;                       // 6250
    k_gemm<<<(tiles + 3) / 4, 128, 0, stream>>>(x, W, dinv, h, out);

    const unsigned scat_threads = (unsigned)NE * 16u;
    k_scatter<<<(scat_threads + 255) / 256, 256, 0, stream>>>(src, dstp, dinv, h, out);

    k_stats   <<<STATS_BLOCKS, 256, 0, stream>>>(out, b, partials);
    k_finalize<<<1, 64, 0, stream>>>(partials, gamma, beta, sshift);
    k_apply   <<<(NN * DIM + 255) / 256, 256, 0, stream>>>(b, sshift, out);
}